// GIN_18846316494850
// MI455X (gfx1250) — compile-verified
//
#include <hip/hip_runtime.h>

typedef float v2f __attribute__((ext_vector_type(2)));
typedef float v8f __attribute__((ext_vector_type(8)));

// ---------------------------------------------------------------------------
// Edge scatter-add: agg[dst[e]] += x[src[e]]  (64 floats per edge)
// One thread handles a 4-float chunk of one edge. L2-resident atomics.
// ---------------------------------------------------------------------------
__global__ __launch_bounds__(256) void gin_scatter_kernel(
    const float* __restrict__ xin, const int* __restrict__ src,
    const int* __restrict__ dst, float* __restrict__ agg, int nEdges)
{
    int tid  = blockIdx.x * 256 + threadIdx.x;
    int e    = tid >> 4;
    if (e >= nEdges) return;
    int c    = (tid & 15) * 4;
    int s    = src[e];
    int d    = dst[e];
    const float4 v = *(const float4*)(xin + (size_t)s * 64 + c);
    float* o = agg + (size_t)d * 64 + c;
    atomicAdd(o + 0, v.x);
    atomicAdd(o + 1, v.y);
    atomicAdd(o + 2, v.z);
    atomicAdd(o + 3, v.w);
}

// ---------------------------------------------------------------------------
// Fused GIN MLP:  out = relu( relu((x+agg) @ Wa + ba) @ Wb + bb )
// One wave32 computes a 16-row tile using V_WMMA_F32_16X16X4_F32 (fp32 WMMA).
//   A frag (16x4 f32):  lanes 0-15 rows, VGPR0/1 = K {4k+koff, 4k+koff+1},
//                       koff = 0 (lanes 0-15) / 2 (lanes 16-31)
//   B frag (4x16 f32):  same K split; N = lane&15
//   C/D (16x16 f32):    VGPR r -> M = r + 8*(lane>=16), N = lane&15
// Intermediate 16x64 tile staged in the wave's private LDS slice.
// ---------------------------------------------------------------------------
__global__ __launch_bounds__(256) void gin_mlp_kernel(
    const float* __restrict__ xin, const float* __restrict__ agg,
    const float* __restrict__ wa,  const float* __restrict__ ba,
    const float* __restrict__ wb,  const float* __restrict__ bb,
    float* __restrict__ out, int nRows, int nTiles)
{
    __shared__ float lds[8 * 16 * 64];               // 32 KB, 4 KB per wave
    const int lane = threadIdx.x & 31;
    const int wave = threadIdx.x >> 5;
    const int tile = blockIdx.x * 8 + wave;
    if (tile >= nTiles) return;                      // wave-uniform exit

    const int row0 = tile * 16;
    const int m    = lane & 15;                      // local row for A frags
    const int koff = (lane >> 4) << 1;               // 0 or 2
    const int colb = lane & 15;                      // local col for B/C frags
    const int rhi  = (lane >> 4) << 3;               // +8 row offset for C frags
    float* wlds    = &lds[wave * 16 * 64];

    // ---- A fragments for GEMM1: (x + agg) row of this lane
    v2f afrag[16];
    {
        int r = row0 + m; if (r >= nRows) r = nRows - 1;   // clamp, no divergence
        const float* xr = xin + (size_t)r * 64;
        const float* gr = agg + (size_t)r * 64;
        #pragma unroll
        for (int k = 0; k < 16; ++k) {
            int c  = 4 * k + koff;
            v2f xa = *(const v2f*)(xr + c);
            v2f ga = *(const v2f*)(gr + c);
            afrag[k] = xa + ga;
        }
    }

    // ---- GEMM1 + bias + ReLU -> LDS (row-major 16x64 tile)
    #pragma unroll
    for (int n = 0; n < 4; ++n) {
        float bias = ba[16 * n + colb];
        v8f acc;
        #pragma unroll
        for (int r = 0; r < 8; ++r) acc[r] = bias;
        #pragma unroll
        for (int k = 0; k < 16; ++k) {
            v2f b;
            b.x = wa[(4 * k + koff)     * 64 + 16 * n + colb];
            b.y = wa[(4 * k + koff + 1) * 64 + 16 * n + colb];
            acc = __builtin_amdgcn_wmma_f32_16x16x4_f32(
                false, afrag[k], false, b, (short)0, acc, false, false);
        }
        #pragma unroll
        for (int r = 0; r < 8; ++r) {
            float v = acc[r] > 0.0f ? acc[r] : 0.0f;
            wlds[(r + rhi) * 64 + 16 * n + colb] = v;
        }
    }

    // ---- A fragments for GEMM2 from LDS (same-wave DS ops are in-order)
    #pragma unroll
    for (int k = 0; k < 16; ++k) {
        int c = 4 * k + koff;
        afrag[k] = *(const v2f*)(wlds + m * 64 + c);
    }

    // ---- GEMM2 + bias + ReLU -> global
    #pragma unroll
    for (int n = 0; n < 4; ++n) {
        float bias = bb[16 * n + colb];
        v8f acc;
        #pragma unroll
        for (int r = 0; r < 8; ++r) acc[r] = bias;
        #pragma unroll
        for (int k = 0; k < 16; ++k) {
            v2f b;
            b.x = wb[(4 * k + koff)     * 64 + 16 * n + colb];
            b.y = wb[(4 * k + koff + 1) * 64 + 16 * n + colb];
            acc = __builtin_amdgcn_wmma_f32_16x16x4_f32(
                false, afrag[k], false, b, (short)0, acc, false, false);
        }
        #pragma unroll
        for (int r = 0; r < 8; ++r) {
            int row = row0 + r + rhi;
            if (row < nRows) {
                float v = acc[r] > 0.0f ? acc[r] : 0.0f;
                out[(size_t)row * 64 + 16 * n + colb] = v;
            }
        }
    }
}

// ---------------------------------------------------------------------------
// Mean-pool phase 1: per-graph feature sums + counts via atomics
// ---------------------------------------------------------------------------
__global__ __launch_bounds__(256) void gin_pool_kernel(
    const float* __restrict__ h, const int* __restrict__ batch,
    float* __restrict__ sums, float* __restrict__ cnts, int nRows)
{
    int tid  = blockIdx.x * 256 + threadIdx.x;
    int node = tid >> 4;
    if (node >= nRows) return;
    int c = (tid & 15) * 4;
    int g = batch[node];
    const float4 v = *(const float4*)(h + (size_t)node * 64 + c);
    float* o = sums + (size_t)g * 64 + c;
    atomicAdd(o + 0, v.x);
    atomicAdd(o + 1, v.y);
    atomicAdd(o + 2, v.z);
    atomicAdd(o + 3, v.w);
    if ((tid & 15) == 0) atomicAdd(&cnts[g], 1.0f);
}

// ---------------------------------------------------------------------------
// Classifier: out[g][c] = bc[c] + (sums[g] @ wc[:,c]) / max(cnt[g],1)
// ---------------------------------------------------------------------------
__global__ __launch_bounds__(256) void gin_cls_kernel(
    const float* __restrict__ sums, const float* __restrict__ cnts,
    const float* __restrict__ wc, const float* __restrict__ bc,
    float* __restrict__ outp, int G, int C)
{
    int tid = blockIdx.x * 256 + threadIdx.x;
    if (tid >= G * C) return;
    int g = tid / C, c = tid % C;
    float cnt = cnts[g];
    if (cnt < 1.0f) cnt = 1.0f;
    float s = 0.0f;
    #pragma unroll 8
    for (int k = 0; k < 64; ++k) s += sums[g * 64 + k] * wc[k * C + c];
    outp[tid] = bc[c] + s / cnt;
}

// ---------------------------------------------------------------------------
extern "C" void kernel_launch(void* const* d_in, const int* in_sizes, int n_in,
                              void* d_out, int out_size, void* d_ws, size_t ws_size,
                              hipStream_t stream)
{
    const float* x   = (const float*)d_in[0];
    const float* w1a = (const float*)d_in[1];
    const float* b1a = (const float*)d_in[2];
    const float* w1b = (const float*)d_in[3];
    const float* b1b = (const float*)d_in[4];
    const float* w2a = (const float*)d_in[5];
    const float* b2a = (const float*)d_in[6];
    const float* w2b = (const float*)d_in[7];
    const float* b2b = (const float*)d_in[8];
    const float* w3a = (const float*)d_in[9];
    const float* b3a = (const float*)d_in[10];
    const float* w3b = (const float*)d_in[11];
    const float* b3b = (const float*)d_in[12];
    const float* wc  = (const float*)d_in[13];
    const float* bc  = (const float*)d_in[14];
    const int*   ei  = (const int*)d_in[15];
    const int*   bat = (const int*)d_in[16];
    float*       out = (float*)d_out;

    const int N = in_sizes[0] / 64;        // 100000
    const int E = in_sizes[15] / 2;        // 1200000
    const int C = 10;
    const int G = out_size / C;            // 128
    const int* src = ei;
    const int* dst = ei + E;

    const size_t NH = (size_t)N * 64;
    float* agg  = (float*)d_ws;            // N*64
    float* bufA = agg  + NH;               // N*64
    float* bufB = bufA + NH;               // N*64
    float* sums = bufB + NH;               // G*64
    float* cnts = sums + (size_t)G * 64;   // G

    const int nTiles     = (N + 15) / 16;
    const int mlpBlocks  = (nTiles + 7) / 8;
    const int scatBlocks = (E * 16 + 255) / 256;
    const int poolBlocks = (N * 16 + 255) / 256;

    // ---- layer 1
    hipMemsetAsync(agg, 0, NH * sizeof(float), stream);
    gin_scatter_kernel<<<scatBlocks, 256, 0, stream>>>(x, src, dst, agg, E);
    gin_mlp_kernel<<<mlpBlocks, 256, 0, stream>>>(x, agg, w1a, b1a, w1b, b1b,
                                                  bufA, N, nTiles);
    // ---- layer 2
    hipMemsetAsync(agg, 0, NH * sizeof(float), stream);
    gin_scatter_kernel<<<scatBlocks, 256, 0, stream>>>(bufA, src, dst, agg, E);
    gin_mlp_kernel<<<mlpBlocks, 256, 0, stream>>>(bufA, agg, w2a, b2a, w2b, b2b,
                                                  bufB, N, nTiles);
    // ---- layer 3
    hipMemsetAsync(agg, 0, NH * sizeof(float), stream);
    gin_scatter_kernel<<<scatBlocks, 256, 0, stream>>>(bufB, src, dst, agg, E);
    gin_mlp_kernel<<<mlpBlocks, 256, 0, stream>>>(bufB, agg, w3a, b3a, w3b, b3b,
                                                  bufA, N, nTiles);

    // ---- pooling + classifier
    hipMemsetAsync(sums, 0, ((size_t)G * 64 + G) * sizeof(float), stream);
    gin_pool_kernel<<<poolBlocks, 256, 0, stream>>>(bufA, bat, sums, cnts, N);
    gin_cls_kernel<<<(G * C + 255) / 256, 256, 0, stream>>>(sums, cnts, wc, bc,
                                                            out, G, C);
}